// MoEBalancedLayer_66718021976460
// MI455X (gfx1250) — compile-verified
//
#include <hip/hip_runtime.h>
#include <hip/hip_bf16.h>
#include <math.h>

typedef float v2f __attribute__((ext_vector_type(2)));
typedef float v4f __attribute__((ext_vector_type(4)));
typedef float v8f __attribute__((ext_vector_type(8)));

#define N_TOK   8192
#define DM      1024
#define DFF     4096
#define NE      8
#define LB_COEF 0.01f
#define BM      32          // tokens per block (2 WMMA row tiles)

// ---------------------------------------------------------------------------
// CDNA5 full-precision matrix op: D = A(16x4,f32) * B(4x16,f32) + C(16x16,f32)
// ---------------------------------------------------------------------------
__device__ __forceinline__ v8f wmma4(v2f a, v2f b, v8f c) {
  return __builtin_amdgcn_wmma_f32_16x16x4_f32(
      /*neg_a=*/false, a, /*neg_b=*/false, b,
      /*c_mod=*/(short)0, c, /*reuse_a=*/false, /*reuse_b=*/false);
}

// ---------------------------------------------------------------------------
__global__ void zero_out_kernel(float* __restrict__ out) {
  size_t i = (size_t)blockIdx.x * blockDim.x + threadIdx.x;
  v4f z = {0.f, 0.f, 0.f, 0.f};
  ((v4f*)out)[i] = z;
}

__global__ void zero_meta_kernel(int* __restrict__ counts) {
  if (threadIdx.x < NE) counts[threadIdx.x] = 0;
}

// ---------------------------------------------------------------------------
// Router: one wave per token. logits = x . gate_w[e], softmax, top-2,
// renormalized combine weights. Deterministic per-block partials for aux.
// grid = 128 blocks x 256 threads; 64 tokens / block, 8 / wave.
// ---------------------------------------------------------------------------
__global__ __launch_bounds__(256) void router_kernel(
    const float* __restrict__ x, const float* __restrict__ gw,
    int* __restrict__ topi, float* __restrict__ topw,
    float* __restrict__ blockP) {
  __shared__ float glds[NE * DM];          // 32 KB
  __shared__ float pW[8][NE];
  __shared__ float fW[8][NE];
  int tid = threadIdx.x, lane = tid & 31, w = tid >> 5;
  for (int i = tid; i < NE * DM; i += 256) glds[i] = gw[i];
  float pAcc[NE], fAcc[NE];
  for (int e = 0; e < NE; ++e) { pAcc[e] = 0.f; fAcc[e] = 0.f; }
  __syncthreads();

  int t0 = blockIdx.x * 64 + w * 8;
  for (int it = 0; it < 8; ++it) {
    int t = t0 + it;
    const float* xr = x + (size_t)t * DM;
    float acc[NE];
    #pragma unroll
    for (int e = 0; e < NE; ++e) acc[e] = 0.f;
    for (int d = lane; d < DM; d += 32) {
      float xv = xr[d];
      #pragma unroll
      for (int e = 0; e < NE; ++e) acc[e] += xv * glds[e * DM + d];
    }
    #pragma unroll
    for (int e = 0; e < NE; ++e)
      for (int off = 16; off; off >>= 1) acc[e] += __shfl_xor(acc[e], off, 32);

    if (lane == 0) {
      float mx = acc[0];
      for (int e = 1; e < NE; ++e) mx = fmaxf(mx, acc[e]);
      float p[NE], s = 0.f;
      for (int e = 0; e < NE; ++e) { p[e] = expf(acc[e] - mx); s += p[e]; }
      float inv = 1.f / s;
      for (int e = 0; e < NE; ++e) p[e] *= inv;
      // top-1 / top-2: first occurrence wins ties (strict >)
      int i1 = 0;
      for (int e = 1; e < NE; ++e) if (p[e] > p[i1]) i1 = e;
      int i2 = (i1 == 0) ? 1 : 0;
      for (int e = 0; e < NE; ++e) if (e != i1 && p[e] > p[i2]) i2 = e;
      float w1v = p[i1], w2v = p[i2], wsum = w1v + w2v;
      topi[t * 2 + 0] = i1;
      topi[t * 2 + 1] = i2;
      topw[t * 2 + 0] = w1v / wsum;
      topw[t * 2 + 1] = w2v / wsum;
      for (int e = 0; e < NE; ++e) pAcc[e] += p[e];
      fAcc[i1] += 1.f;
    }
  }
  if (lane == 0)
    for (int e = 0; e < NE; ++e) { pW[w][e] = pAcc[e]; fW[w][e] = fAcc[e]; }
  __syncthreads();
  if (tid < NE) {
    float ps = 0.f, fs = 0.f;
    for (int ww = 0; ww < 8; ++ww) { ps += pW[ww][tid]; fs += fW[ww][tid]; }
    blockP[blockIdx.x * 2 * NE + tid]      = ps;
    blockP[blockIdx.x * 2 * NE + NE + tid] = fs;
  }
}

// Fixed-order reduction of router partials -> aux loss scalar.
__global__ void aux_kernel(const float* __restrict__ blockP,
                           float* __restrict__ auxOut, int nblocks) {
  __shared__ float pS[NE], fS[NE];
  int e = threadIdx.x;
  if (e < NE) {
    float ps = 0.f, fs = 0.f;
    for (int b = 0; b < nblocks; ++b) {
      ps += blockP[b * 2 * NE + e];
      fs += blockP[b * 2 * NE + NE + e];
    }
    pS[e] = ps; fS[e] = fs;
  }
  __syncthreads();
  if (e == 0) {
    float a = 0.f;
    for (int k = 0; k < NE; ++k)
      a += (fS[k] / (float)N_TOK) * (pS[k] / (float)N_TOK);
    auxOut[0] = LB_COEF * a;
  }
}

__global__ void count_kernel(const int* __restrict__ topi, int* __restrict__ counts) {
  int i = blockIdx.x * blockDim.x + threadIdx.x;   // 2N entries
  atomicAdd(&counts[topi[i]], 1);
}

__global__ void prefix_kernel(const int* __restrict__ counts,
                              int* __restrict__ basep, int* __restrict__ cursor) {
  if (threadIdx.x == 0) {
    int acc = 0;
    for (int e = 0; e < NE; ++e) { basep[e] = acc; cursor[e] = acc; acc += counts[e]; }
  }
}

__global__ void scatter_kernel(const int* __restrict__ topi, const float* __restrict__ topw,
                               int* __restrict__ cursor,
                               int* __restrict__ tokS, float* __restrict__ wS) {
  int i = blockIdx.x * blockDim.x + threadIdx.x;   // 2N entries
  int e = topi[i];
  int pos = atomicAdd(&cursor[e], 1);
  tokS[pos] = i >> 1;
  wS[pos]  = topw[i];
}

// ---------------------------------------------------------------------------
// Fused expert FFN: y = gelu(X @ W1 + b1) @ W2 + b2, scaled & scattered.
// Block = 32 routed tokens of one expert, 256 threads = 8 waves.
// LDS (packed K-pair layout): element pair p of row r lives at
//   ((v2f*)buf)[p*BM + r] = (val[2p], val[2p+1])
// so a WMMA A-fragment (rows x K+0/K+1 | K+2/K+3 per lane half) is ONE
// naturally aligned ds_load_b64 landing in a contiguous even VGPR pair --
// no v_mov fixups in front of the WMMAs.
// FF loop: 32 chunks of 128.
//   GEMM1: wave w -> 2 row-tiles of one 16-col H slice (K=1024); each W1
//          B fragment (2 global dwords) feeds TWO wmma.
//   GEMM2: wave w -> 32x128 y slice = 16 WMMA C tiles (128 VGPRs).
// ---------------------------------------------------------------------------
__global__ __launch_bounds__(256) void moe_ffn_kernel(
    const float* __restrict__ x,
    const float* __restrict__ W1, const float* __restrict__ b1,
    const float* __restrict__ W2, const float* __restrict__ b2,
    const int* __restrict__ counts, const int* __restrict__ basep,
    const int* __restrict__ tokS, const float* __restrict__ wS,
    float* __restrict__ out) {
  int e = blockIdx.y;
  int cnt = counts[e];
  int row0 = blockIdx.x * BM;
  if (row0 >= cnt) return;                      // uniform block-level exit

  extern __shared__ float smem[];
  float* Xt = smem;                             // packed pairs: [DM/2][BM] v2f
  float* Hs = smem + DM * BM;                   // packed pairs: [64][BM] v2f
  int*   tokLds = (int*)(Hs + 128 * BM);        // [BM]
  float* wLds   = (float*)(tokLds + BM);        // [BM]

  const float* W1e = W1 + (size_t)e * DM * DFF;
  const float* W2e = W2 + (size_t)e * DFF * DM;
  const float* b1e = b1 + (size_t)e * DFF;
  const float* b2e = b2 + (size_t)e * DM;

  int tid = threadIdx.x, lane = tid & 31, w = tid >> 5;
  int bbase = basep[e];
  if (tid < BM) {
    int r = row0 + tid;
    tokLds[tid] = (r < cnt) ? tokS[bbase + r] : 0;
    wLds[tid]   = (r < cnt) ? wS[bbase + r] : 0.f;
  }
  __syncthreads();

  // Gather X rows (coalesced along d); packed-pair transposed store:
  // Xt float index = (d>>1)*2*BM + r*2 + (d&1)
  for (int idx = tid; idx < BM * DM; idx += 256) {
    int r = idx >> 10;
    int d = idx & (DM - 1);
    Xt[(d >> 1) * (2 * BM) + r * 2 + (d & 1)] = x[(size_t)tokLds[r] * DM + d];
  }
  __syncthreads();

  int half = (lane >= 16) ? 1 : 0;
  int ln   = lane & 15;
  int krow = half * 2;

  const v2f* Xp = (const v2f*)Xt;
  const v2f* Hp = (const v2f*)Hs;

  v8f acc[16];
  #pragma unroll
  for (int t = 0; t < 16; ++t) acc[t] = (v8f){0, 0, 0, 0, 0, 0, 0, 0};

  for (int c = 0; c < DFF / 128; ++c) {
    int n0 = c * 128;
    // ---- GEMM1: two 16x16 H tiles, K = 1024 ----------------------------
    v8f hc0 = (v8f){0, 0, 0, 0, 0, 0, 0, 0};
    v8f hc1 = (v8f){0, 0, 0, 0, 0, 0, 0, 0};
    const float* w1p = W1e + (size_t)krow * DFF + (n0 + w * 16 + ln);
    const v2f*   xq  = Xp + half * BM + ln;      // + (k/2)*BM per step
    #pragma unroll 4
    for (int k = 0; k < DM; k += 4) {
      v2f b;
      b.x = w1p[(size_t)k * DFF];
      b.y = w1p[(size_t)k * DFF + DFF];
      v2f a0 = xq[(k >> 1) * BM];
      v2f a1 = xq[(k >> 1) * BM + 16];
      hc0 = wmma4(a0, b, hc0);
      hc1 = wmma4(a1, b, hc1);
    }
    // bias + exact GELU, packed-pair store of H^T (column nn = w*16+ln):
    // float index = (nn>>1)*2*BM + m*2 + (nn&1)
    float b1v = b1e[n0 + w * 16 + ln];
    int nn = w * 16 + ln;
    int hbase = (nn >> 1) * (2 * BM) + (nn & 1);
    #pragma unroll
    for (int i = 0; i < 8; ++i) {
      int m0 = i + half * 8;
      float v0 = hc0[i] + b1v;
      v0 = 0.5f * v0 * (1.0f + erff(v0 * 0.70710678118654752f));
      Hs[hbase + m0 * 2] = v0;
      float v1 = hc1[i] + b1v;
      v1 = 0.5f * v1 * (1.0f + erff(v1 * 0.70710678118654752f));
      Hs[hbase + (m0 + 16) * 2] = v1;
    }
    __syncthreads();
    // ---- GEMM2: y slice (32x128) += H(32x128) @ W2 chunk ---------------
    const float* w2p = W2e + (size_t)(n0 + krow) * DM + (w * 128 + ln);
    const v2f*   hq  = Hp + half * BM + ln;      // + (k/2)*BM per step
    #pragma unroll 2
    for (int k = 0; k < 128; k += 4) {
      v2f a0 = hq[(k >> 1) * BM];
      v2f a1 = hq[(k >> 1) * BM + 16];
      const float* w2k = w2p + (size_t)k * DM;
      #pragma unroll
      for (int t = 0; t < 8; ++t) {
        v2f b;
        b.x = w2k[t * 16];
        b.y = w2k[t * 16 + DM];
        acc[t]     = wmma4(a0, b, acc[t]);
        acc[8 + t] = wmma4(a1, b, acc[8 + t]);
      }
    }
    __syncthreads();
  }

  // Epilogue: out[token] += weight * (y + b2). Two commutative f32 atomic
  // adds per element (one per routed expert) -> value-deterministic.
  #pragma unroll
  for (int t = 0; t < 8; ++t) {
    int col = w * 128 + t * 16 + ln;
    float b2v = b2e[col];
    #pragma unroll
    for (int rt = 0; rt < 2; ++rt) {
      #pragma unroll
      for (int i = 0; i < 8; ++i) {
        int m = i + half * 8 + rt * 16;
        if (row0 + m < cnt) {
          float val = wLds[m] * (acc[rt * 8 + t][i] + b2v);
          atomicAdd(&out[(size_t)tokLds[m] * DM + col], val);
        }
      }
    }
  }
}

// ---------------------------------------------------------------------------
extern "C" void kernel_launch(void* const* d_in, const int* in_sizes, int n_in,
                              void* d_out, int out_size, void* d_ws, size_t ws_size,
                              hipStream_t stream) {
  const float* x  = (const float*)d_in[0];
  const float* gw = (const float*)d_in[1];
  const float* W1 = (const float*)d_in[2];
  const float* b1 = (const float*)d_in[3];
  const float* W2 = (const float*)d_in[4];
  const float* b2 = (const float*)d_in[5];
  float* out = (float*)d_out;

  char* p = (char*)d_ws;
  int*   topi   = (int*)p;    p += (size_t)N_TOK * 2 * sizeof(int);
  float* topw   = (float*)p;  p += (size_t)N_TOK * 2 * sizeof(float);
  int*   counts = (int*)p;    p += 16 * sizeof(int);
  int*   basep  = (int*)p;    p += 16 * sizeof(int);
  int*   cursor = (int*)p;    p += 16 * sizeof(int);
  float* blockP = (float*)p;  p += (size_t)128 * 2 * NE * sizeof(float);
  int*   tokS   = (int*)p;    p += (size_t)N_TOK * 2 * sizeof(int);
  float* wSrt   = (float*)p;  p += (size_t)N_TOK * 2 * sizeof(float);

  zero_out_kernel<<<dim3((N_TOK * DM) / (256 * 4)), dim3(256), 0, stream>>>(out);
  zero_meta_kernel<<<dim3(1), dim3(64), 0, stream>>>(counts);
  router_kernel<<<dim3(128), dim3(256), 0, stream>>>(x, gw, topi, topw, blockP);
  aux_kernel<<<dim3(1), dim3(NE), 0, stream>>>(blockP, out + (size_t)N_TOK * DM, 128);
  count_kernel<<<dim3(64), dim3(256), 0, stream>>>(topi, counts);
  prefix_kernel<<<dim3(1), dim3(1), 0, stream>>>(counts, basep, cursor);
  scatter_kernel<<<dim3(64), dim3(256), 0, stream>>>(topi, topw, cursor, tokS, wSrt);

  size_t shmem = (size_t)(DM * BM + 128 * BM) * sizeof(float)
               + BM * sizeof(int) + BM * sizeof(float);
  moe_ffn_kernel<<<dim3(N_TOK / BM, NE), dim3(256), shmem, stream>>>(
      x, W1, b1, W2, b2, counts, basep, tokS, wSrt, out);
}